// SelfAttentionRVPooling_45535243272771
// MI455X (gfx1250) — compile-verified
//
#include <hip/hip_runtime.h>

// Problem constants from the reference
#define NPG 512      // nodes per graph
#define KSEL 256     // nodes kept per graph (RATIO 0.5)
#define FDIM 255     // raw feature dim (x has FDIM+1 = 256 cols, last = real bit)

typedef __attribute__((ext_vector_type(2))) float v2f;
typedef __attribute__((ext_vector_type(8))) float v8f;

// ---------------------------------------------------------------------------
// h[n] = x[n,:] . W  via V_WMMA_F32_16X16X4_F32
// A = 16 nodes x 4 features (lanes 0-15: K=0,1 ; lanes 16-31: K=2,3 per ISA
// layout). B = W[k..k+3] broadcast across all 16 columns, so every column of
// the 16x16 f32 accumulator holds h for its row. 64 WMMAs cover K=256.
// blockDim = 256 (8 waves), each wave owns 16 nodes.
// ---------------------------------------------------------------------------
__global__ void k_score(const float* __restrict__ xf, const int* __restrict__ z,
                        const float* __restrict__ W, float* __restrict__ h) {
  const int lane = threadIdx.x & 31;
  const int wave = threadIdx.x >> 5;
  const int nodeBase = blockIdx.x * 128 + wave * 16;
  const int m = lane & 15;
  const int node = nodeBase + m;
  const int koff = (lane >> 4) << 1;                 // 0 for lanes 0-15, 2 for 16-31
  const long rowOff = (long)node * FDIM;

  v8f c = {0.f, 0.f, 0.f, 0.f, 0.f, 0.f, 0.f, 0.f};
  v2f a, bv;

#pragma unroll 4
  for (int kk = 0; kk < 252; kk += 4) {
    a.x  = xf[rowOff + kk + koff];
    a.y  = xf[rowOff + kk + koff + 1];
    bv.x = W[kk + koff];
    bv.y = W[kk + koff + 1];
    c = __builtin_amdgcn_wmma_f32_16x16x4_f32(false, a, false, bv,
                                              (short)0, c, false, false);
  }
  // Tail: features 252,253,254 from memory; feature 255 is the real-node bit.
  // Branchless (cndmask) so EXEC stays all-1s for the WMMA.
  {
    const float realbit = (z[node] != 100) ? 1.0f : 0.0f;
    const float f253 = xf[rowOff + 253];             // all lanes load; hi lanes discard
    a.x  = xf[rowOff + 252 + koff];                  // 252 (lo) or 254 (hi)
    a.y  = (koff != 0) ? realbit : f253;
    bv.x = W[252 + koff];
    bv.y = W[253 + koff];                            // W[253] or W[255]
    c = __builtin_amdgcn_wmma_f32_16x16x4_f32(false, a, false, bv,
                                              (short)0, c, false, false);
  }
  // C layout: VGPR j holds M=j (lanes 0-15) and M=j+8 (lanes 16-31); all 16
  // columns are identical here, so lane 0 / lane 16 each emit 8 node scores.
  if (lane == 0) {
#pragma unroll
    for (int j = 0; j < 8; ++j) h[nodeBase + j] = c[j];
  }
  if (lane == 16) {
#pragma unroll
    for (int j = 0; j < 8; ++j) h[nodeBase + 8 + j] = c[j];
  }
}

// deg starts at 1 (self-loop)
__global__ void k_initdeg(unsigned* __restrict__ deg) {
  deg[blockIdx.x * blockDim.x + threadIdx.x] = 1u;
}

// in-degree over edge cols
__global__ void k_deg(const int* __restrict__ ei, unsigned* __restrict__ deg,
                      const int E) {
  const int e = blockIdx.x * blockDim.x + threadIdx.x;
  if (e < E) atomicAdd(&deg[ei[E + e]], 1u);
}

// dinv = rsqrt(deg); agg seeded with self-loop term dinv^2 * h
__global__ void k_dinv(const unsigned* __restrict__ deg,
                       const float* __restrict__ h,
                       float* __restrict__ dinv, float* __restrict__ agg) {
  const int n = blockIdx.x * blockDim.x + threadIdx.x;
  const float di = rsqrtf((float)deg[n]);
  dinv[n] = di;
  agg[n] = di * di * h[n];
}

// agg[col] += dinv[row]*dinv[col]*h[row]
__global__ void k_edge(const int* __restrict__ ei, const float* __restrict__ dinv,
                       const float* __restrict__ h, float* __restrict__ agg,
                       const int E) {
  const int e = blockIdx.x * blockDim.x + threadIdx.x;
  if (e < E) {
    const int r = ei[e];
    const int cidx = ei[E + e];
    atomicAdd(&agg[cidx], dinv[r] * dinv[cidx] * h[r]);
  }
}

// Per-graph top-K by exact ranking (value desc, index asc): rank is a
// permutation of 0..NPG-1, kept iff rank < KSEL, and rank is the output slot
// (deterministic, matches jax.lax.top_k tie behavior; order irrelevant for
// the later mean anyway).
__global__ void k_topk(const float* __restrict__ agg, const float* __restrict__ b,
                       int* __restrict__ sel_idx, float* __restrict__ sel_scr) {
  __shared__ float s[NPG];
  const int g = blockIdx.x;
  const float bb = b[0];
  for (int i = threadIdx.x; i < NPG; i += blockDim.x)
    s[i] = tanhf(agg[g * NPG + i] + bb);
  __syncthreads();
  for (int i = threadIdx.x; i < NPG; i += blockDim.x) {
    const float si = s[i];
    int rank = 0;
    for (int j = 0; j < NPG; ++j) {
      const float sj = s[j];
      rank += (sj > si) || ((sj == si) && (j < i)) ? 1 : 0;
    }
    if (rank < KSEL) {
      sel_idx[g * KSEL + rank] = g * NPG + i;   // global node id
      sel_scr[g * KSEL + rank] = si;
    }
  }
}

// out[g,f] = (1/K) * sum_k scr[k] * x[sel[k], f]  (f = 255 -> real bit)
// One block per graph; thread t owns feature f=t; per-k loads are coalesced
// across the 255 feature lanes.
__global__ void k_pool(const float* __restrict__ xf, const int* __restrict__ z,
                       const int* __restrict__ sel_idx,
                       const float* __restrict__ sel_scr,
                       float* __restrict__ dout) {
  __shared__ int   li[KSEL];
  __shared__ float ls[KSEL];
  const int g = blockIdx.x;
  const int f = threadIdx.x;
  li[f] = sel_idx[g * KSEL + f];
  ls[f] = sel_scr[g * KSEL + f];
  __syncthreads();
  float acc = 0.0f;
  if (f < FDIM) {
    for (int k = 0; k < KSEL; ++k)
      acc += ls[k] * xf[(long)li[k] * FDIM + f];
  } else {
    for (int k = 0; k < KSEL; ++k)
      acc += ls[k] * ((z[li[k]] != 100) ? 1.0f : 0.0f);
  }
  dout[g * (FDIM + 1) + f] = acc * (1.0f / (float)KSEL);
}

// ---------------------------------------------------------------------------
extern "C" void kernel_launch(void* const* d_in, const int* in_sizes, int n_in,
                              void* d_out, int out_size, void* d_ws, size_t ws_size,
                              hipStream_t stream) {
  const float* xf = (const float*)d_in[0];   // out: [N, 255] f32
  const int*   z  = (const int*)d_in[1];     // z:   [N] i32
  const int*   ei = (const int*)d_in[2];     // edge_index: [2, E] i32
  const float* W  = (const float*)d_in[4];   // [256, 1] f32
  const float* b  = (const float*)d_in[5];   // [1] f32

  const int N = in_sizes[1];                 // 262144
  const int E = in_sizes[2] / 2;             // 8388608
  const int B = N / NPG;                     // 512

  char* ws = (char*)d_ws;
  float*    h       = (float*)(ws);
  float*    dinv    = (float*)(ws + 4ull * N);
  float*    agg     = (float*)(ws + 8ull * N);
  unsigned* deg     = (unsigned*)(ws + 12ull * N);
  int*      sel_idx = (int*)(ws + 16ull * N);
  float*    sel_scr = (float*)(ws + 16ull * N + 4ull * (size_t)B * KSEL);
  float*    dout    = (float*)d_out;

  k_score  <<<N / 128, 256, 0, stream>>>(xf, z, W, h);
  k_initdeg<<<N / 256, 256, 0, stream>>>(deg);
  k_deg    <<<(E + 255) / 256, 256, 0, stream>>>(ei, deg, E);
  k_dinv   <<<N / 256, 256, 0, stream>>>(deg, h, dinv, agg);
  k_edge   <<<(E + 255) / 256, 256, 0, stream>>>(ei, dinv, h, agg, E);
  k_topk   <<<B, 256, 0, stream>>>(agg, b, sel_idx, sel_scr);
  k_pool   <<<B, 256, 0, stream>>>(xf, z, sel_idx, sel_scr, dout);
}